// BimodalCSRPool_55946243997763
// MI455X (gfx1250) — compile-verified
//
#include <hip/hip_runtime.h>

// BimodalCSRPool: segment-CSR max over x_mod rows (D=128 f32), empty -> 0,
// plus x_seen flag. Pure HBM-bandwidth problem (~461 MB => ~20us @ 23.3 TB/s).
// Mapping: one wave32 per segment; lane l owns float4 channels [4l, 4l+4).
// One row = 32 lanes x b128 = 512B coalesced; a segment's rows are contiguous.

typedef float f32x4 __attribute__((ext_vector_type(4)));

__global__ __launch_bounds__(256) void
bimodal_csr_pool_max(const float* __restrict__ xmod,
                     const int*   __restrict__ csr,
                     float*       __restrict__ pool,
                     float*       __restrict__ seen,
                     int nseg)
{
    const int gtid = blockIdx.x * 256 + threadIdx.x;
    const int lane = threadIdx.x & 31;

    // wave-uniform segment id: force SGPR so csr reads go down the scalar
    // (s_load_b64) path instead of per-lane VMEM.
    const int seg = __builtin_amdgcn_readfirstlane(gtid >> 5);
    if (seg >= nseg) return;

    const int beg = csr[seg];
    const int end = csr[seg + 1];
    const int rows = end - beg;

    // Run a uniform max loop starting from -inf; empty segments are patched
    // to 0 afterwards (torch_scatter segment_csr semantics).
    f32x4 acc = (f32x4)(-__builtin_inff());

    // row r starts at xmod + (beg+r)*128 floats = (f32x4*)xmod + (beg+r)*32
    const f32x4* p = (const f32x4*)xmod + (long)beg * 32 + lane;

    int r = 0;
    // 4-deep unroll: 4 global_load_b128 in flight per wave before the wait.
    for (; r + 4 <= rows; r += 4) {
        __builtin_prefetch((const void*)(p + 128), 0, 0);   // global_prefetch one block ahead
        f32x4 a = __builtin_nontemporal_load(p);
        f32x4 b = __builtin_nontemporal_load(p + 32);
        f32x4 c = __builtin_nontemporal_load(p + 64);
        f32x4 d = __builtin_nontemporal_load(p + 96);
        p += 128;
        f32x4 m0 = __builtin_elementwise_max(a, b);
        f32x4 m1 = __builtin_elementwise_max(c, d);
        acc = __builtin_elementwise_max(acc, __builtin_elementwise_max(m0, m1));
    }
    for (; r < rows; ++r) {
        f32x4 a = __builtin_nontemporal_load(p);
        p += 32;
        acc = __builtin_elementwise_max(acc, a);
    }

    if (rows <= 0)
        acc = (f32x4)(0.0f);

    // pool row for this segment: 32 f32x4, lane-coalesced 512B store
    f32x4* op = (f32x4*)pool + (long)seg * 32 + lane;
    __builtin_nontemporal_store(acc, op);

    if (lane == 0)
        seen[seg] = (rows > 0) ? 1.0f : 0.0f;
}

extern "C" void kernel_launch(void* const* d_in, const int* in_sizes, int n_in,
                              void* d_out, int out_size, void* d_ws, size_t ws_size,
                              hipStream_t stream) {
    // inputs (setup_inputs order): x_main [N*D] f32 (UNUSED by reference),
    //                              x_mod  [E*D] f32,
    //                              csr_idx [N+1] i32
    const float* xmod = (const float*)d_in[1];
    const int*   csr  = (const int*)d_in[2];

    const int nseg = in_sizes[2] - 1;          // N
    float* pool = (float*)d_out;                           // [N, 128]
    float* seen = (float*)d_out + (size_t)nseg * 128;      // [N] as 0.0/1.0

    // one wave32 per segment, 8 waves (256 threads) per block
    const int blocks = (nseg + 7) / 8;
    bimodal_csr_pool_max<<<blocks, 256, 0, stream>>>(xmod, csr, pool, seen, nseg);
}